// Som_79362405695813
// MI455X (gfx1250) — compile-verified
//
#include <hip/hip_runtime.h>

// SOM BMU search on gfx1250 (MI455X).
// argmin_m ||x - w_m||^2 == argmin_m (||w_m||^2 - 2 x.w_m)
// Core: 4096 x 16384 x 64 fp32 GEMM via V_WMMA_F32_16X16X4_F32.
//  - B fragments register-double-buffered (16 batched b64 loads per tile,
//    one deferred s_wait, overlapped with the 2x16 WMMA chain of the
//    previous tile).
//  - Each wave carries TWO 16-row x-slices (2 WMMAs per B fragment) to halve
//    L2 traffic for the codebook.
//  - ||w||^2 fused into the same K loop; argmin fused on top.

typedef __attribute__((ext_vector_type(2))) float v2f;
typedef __attribute__((ext_vector_type(8))) float v8f;

#define SOM_B    4096
#define SOM_ROWS 128
#define SOM_COLS 128
#define SOM_M    (SOM_ROWS * SOM_COLS)   // 16384 codebook entries
#define SOM_F    64                      // feature dim (K)

#define WAVES_PER_BLK 8
#define ROWS_PER_BLK  32                 // 2 x-slices of 16 per wave
#define TILES_TOTAL   (SOM_M / 16)       // 1024 column tiles of 16
#define TILES_PER_WAVE (TILES_TOTAL / WAVES_PER_BLK)  // 128 (even)

#define WMMA_F32(acc, A, Bv) \
    __builtin_amdgcn_wmma_f32_16x16x4_f32(false, (A), false, (Bv), (short)0, (acc), false, false)

__global__ __launch_bounds__(256) void som_bmu_kernel(
    const float* __restrict__ x,   // [4096, 64]
    const float* __restrict__ w,   // [16384, 64] (128*128 flattened)
    float* __restrict__ out)       // [4096, 2]  (row, col) as floats
{
    __shared__ float s_min[WAVES_PER_BLK * ROWS_PER_BLK];
    __shared__ int   s_idx[WAVES_PER_BLK * ROWS_PER_BLK];

    const int wave = threadIdx.x >> 5;
    const int lane = threadIdx.x & 31;
    const int half = lane >> 4;     // which K-half of the fragment this lane holds
    const int ln   = lane & 15;     // row-of-A / col-of-B index within tile
    const int row_base = blockIdx.x * ROWS_PER_BLK;

    // ---- A fragments: two x slices (16 rows x 64 K each) resident in regs.
    // f32 16x4 A layout: lane(l) -> M = l%16 ; VGPR0 = K=2*(l/16), VGPR1 = K+1.
    v2f a0[16], a1[16];
    {
        const float* xr0 = x + (size_t)(row_base + ln) * SOM_F + 2 * half;
        const float* xr1 = x + (size_t)(row_base + 16 + ln) * SOM_F + 2 * half;
#pragma unroll
        for (int s = 0; s < 16; ++s) {
            a0[s] = *(const v2f*)(xr0 + 4 * s);
            a1[s] = *(const v2f*)(xr1 + 4 * s);
        }
    }

    float run_min0[8], run_min1[8];
    int   run_idx0[8], run_idx1[8];
#pragma unroll
    for (int r = 0; r < 8; ++r) {
        run_min0[r] = 3.4e38f; run_idx0[r] = 0;
        run_min1[r] = 3.4e38f; run_idx1[r] = 0;
    }

#define LOAD_TILE(buf, jt)                                                   \
    {                                                                        \
        const float* wr = w + (size_t)((jt) * 16 + ln) * SOM_F + 2 * half;   \
        _Pragma("unroll")                                                    \
        for (int s = 0; s < 16; ++s)                                         \
            (buf)[s] = *(const v2f*)(wr + 4 * s);                            \
    }

#define COMPUTE_TILE(buf, jt)                                                \
    {                                                                        \
        v8f acc0 = {0.f,0.f,0.f,0.f,0.f,0.f,0.f,0.f};                        \
        v8f acc1 = {0.f,0.f,0.f,0.f,0.f,0.f,0.f,0.f};                        \
        float nrm = 0.0f;                                                    \
        _Pragma("unroll")                                                    \
        for (int s = 0; s < 16; ++s) {                                       \
            nrm  = fmaf((buf)[s].x, (buf)[s].x, nrm);                        \
            nrm  = fmaf((buf)[s].y, (buf)[s].y, nrm);                        \
            acc0 = WMMA_F32(acc0, a0[s], (buf)[s]);                          \
            acc1 = WMMA_F32(acc1, a1[s], (buf)[s]);                          \
        }                                                                    \
        nrm += __shfl_xor(nrm, 16, 32);                                      \
        const int cidx = (jt) * 16 + ln;                                     \
        _Pragma("unroll")                                                    \
        for (int r = 0; r < 8; ++r) {                                        \
            float sc0 = fmaf(-2.0f, acc0[r], nrm);                           \
            if (sc0 < run_min0[r]) { run_min0[r] = sc0; run_idx0[r] = cidx; }\
            float sc1 = fmaf(-2.0f, acc1[r], nrm);                           \
            if (sc1 < run_min1[r]) { run_min1[r] = sc1; run_idx1[r] = cidx; }\
        }                                                                    \
    }

    // ---- Main sweep: tiles wave, wave+8, wave+16, ... (interleaved so the
    // block walks W contiguously); double-buffered in b0/b1.
    v2f b0[16], b1[16];
    int jt = wave;                 // tile index of the data currently in b0
    LOAD_TILE(b0, jt);

    for (int i = 0; i < TILES_PER_WAVE / 2; ++i) {
        const int jt1 = jt + WAVES_PER_BLK;

        // HBM-warming prefetch a few tiles ahead (first pass only matters).
        if (i + 2 < TILES_PER_WAVE / 2) {
            const float* pre =
                w + (size_t)((jt + 4 * WAVES_PER_BLK) * 16 + ln) * SOM_F + half * 32;
            __builtin_prefetch(pre, 0, 1);
        }

        LOAD_TILE(b1, jt1);        // loads for tile k+1 in flight ...
        COMPUTE_TILE(b0, jt);      // ... while computing tile k

        const int jt2 = jt + 2 * WAVES_PER_BLK;
        if (i + 1 < TILES_PER_WAVE / 2)
            LOAD_TILE(b0, jt2);    // loads for tile k+2 in flight ...
        COMPUTE_TILE(b1, jt1);     // ... while computing tile k+1

        jt = jt2;
    }

    // ---- Argmin across the 16 lanes of each half (C/D row r+8*half lives
    // spread over lanes of that half). xor masks 1..8 stay within a half.
#pragma unroll
    for (int r = 0; r < 8; ++r) {
        float mn0 = run_min0[r]; int ix0 = run_idx0[r];
        float mn1 = run_min1[r]; int ix1 = run_idx1[r];
#pragma unroll
        for (int d = 1; d < 16; d <<= 1) {
            float o0 = __shfl_xor(mn0, d, 32); int j0 = __shfl_xor(ix0, d, 32);
            if (o0 < mn0 || (o0 == mn0 && j0 < ix0)) { mn0 = o0; ix0 = j0; }
            float o1 = __shfl_xor(mn1, d, 32); int j1 = __shfl_xor(ix1, d, 32);
            if (o1 < mn1 || (o1 == mn1 && j1 < ix1)) { mn1 = o1; ix1 = j1; }
        }
        run_min0[r] = mn0; run_idx0[r] = ix0;
        run_min1[r] = mn1; run_idx1[r] = ix1;
    }

    // Lane 0 of each half holds the half's result for rows r / r+8 (per slice).
    if (ln == 0) {
#pragma unroll
        for (int r = 0; r < 8; ++r) {
            const int m = r + 8 * half;
            s_min[wave * ROWS_PER_BLK + m]      = run_min0[r];
            s_idx[wave * ROWS_PER_BLK + m]      = run_idx0[r];
            s_min[wave * ROWS_PER_BLK + 16 + m] = run_min1[r];
            s_idx[wave * ROWS_PER_BLK + 16 + m] = run_idx1[r];
        }
    }
    __syncthreads();

    // ---- Cross-wave reduce (8 candidates per x-row) + emit BMU grid coords.
    if (threadIdx.x < ROWS_PER_BLK) {
        const int t = threadIdx.x;
        float mn = s_min[t];
        int   ix = s_idx[t];
#pragma unroll
        for (int k = 1; k < WAVES_PER_BLK; ++k) {
            float omn = s_min[k * ROWS_PER_BLK + t];
            int   oix = s_idx[k * ROWS_PER_BLK + t];
            if (omn < mn || (omn == mn && oix < ix)) { mn = omn; ix = oix; }
        }
        const int row = row_base + t;
        out[2 * row + 0] = (float)(ix / SOM_COLS);
        out[2 * row + 1] = (float)(ix % SOM_COLS);
    }
}

extern "C" void kernel_launch(void* const* d_in, const int* in_sizes, int n_in,
                              void* d_out, int out_size, void* d_ws, size_t ws_size,
                              hipStream_t stream) {
    (void)in_sizes; (void)n_in; (void)d_ws; (void)ws_size; (void)out_size;
    const float* x = (const float*)d_in[0];   // [4096, 64]
    const float* w = (const float*)d_in[1];   // [128*128, 64]
    float* out = (float*)d_out;               // [4096, 2]
    som_bmu_kernel<<<SOM_B / ROWS_PER_BLK, 256, 0, stream>>>(x, w, out);
}